// SubwordAggregation_89593017795082
// MI455X (gfx1250) — compile-verified
//
#include <hip/hip_runtime.h>
#include <stdint.h>

#define HDIM 512
#define WAVE 32

// ---------------- workspace layout (int32 units) ----------------
#define WS_SRC_Q   0        // compacted True positions of question_mask_plm
#define WS_SRC_T   12288    // table_mask_plm
#define WS_SRC_C   24576    // column_mask_plm
#define WS_L_Q     36864    // per-word subword count
#define WS_P_Q     40960    // per-word exclusive prefix into SRC list
#define WS_L_T     45056
#define WS_P_T     49152
#define WS_L_C     53248
#define WS_P_C     57344
#define WS_D_Q     61440    // compacted True positions of question_mask
#define WS_D_TW    65536    // table_word_mask
#define WS_D_CW    69632    // column_word_mask
#define WS_D_TT    73728    // table_total_mask
#define WS_D_CT    77824    // column_total_mask
#define WS_META    81920
// meta indices
#define M_NQ 0
#define M_NT 1
#define M_NC 2

// ---------------- wave32 scan/compact helpers ----------------
__device__ __forceinline__ int wave_popcount_mask(const int* m, int n, int lane) {
    int c = 0;
    for (int i = lane; i < n; i += WAVE) c += (m[i] != 0);
    #pragma unroll
    for (int off = 16; off > 0; off >>= 1) c += __shfl_xor(c, off, WAVE);
    return c;
}

// stable compaction: out[r] = flat index of r-th True; returns count
__device__ __forceinline__ int wave_compact(const int* m, int n, int* out, int lane) {
    int total = 0;
    for (int base = 0; base < n; base += WAVE) {
        int i = base + lane;
        int p = (i < n) && (m[i] != 0);
        unsigned bal = (unsigned)__ballot(p);          // wave32: low 32 bits
        int rank = __popc(bal & ((1u << lane) - 1u));
        if (p) out[total + rank] = i;
        total += __popc(bal);
    }
    return total;
}

// per-row length + exclusive prefix for a left-aligned [N,S] mask
__device__ __forceinline__ void word_table(const int* m, int N, int S,
                                           int* L, int* P, int lane) {
    int carry = 0;
    for (int base = 0; base < N; base += WAVE) {
        int n = base + lane;
        int len = 0;
        if (n < N) {
            const int* row = m + (long)n * S;
            for (int k = 0; k < S; ++k) len += (row[k] != 0);
        }
        int scan = len;
        #pragma unroll
        for (int off = 1; off < WAVE; off <<= 1) {
            int v = __shfl_up(scan, off, WAVE);
            if (lane >= off) scan += v;
        }
        if (n < N) { L[n] = len; P[n] = carry + scan - len; }
        carry += __shfl(scan, WAVE - 1, WAVE);
    }
}

// ---------------- kernel 1: zero the output ----------------
__global__ void SubAgg_zero_kernel(float4* __restrict__ out, long n4) {
    long i = (long)blockIdx.x * blockDim.x + threadIdx.x;
    long stride = (long)gridDim.x * blockDim.x;
    float4 z; z.x = 0.f; z.y = 0.f; z.z = 0.f; z.w = 0.f;
    for (; i < n4; i += stride) out[i] = z;
}

// ---------------- kernel 2: build all index tables (11 waves) ----------------
__global__ void SubAgg_index_kernel(
    const int* __restrict__ q_sub,  int s0,
    const int* __restrict__ t_sub,  int s1,
    const int* __restrict__ c_sub,  int s2,
    const int* __restrict__ q_mask, int s3,
    const int* __restrict__ tw_mask, int s4,
    const int* __restrict__ cw_mask, int s5,
    const int* __restrict__ tt_mask, int s6,
    const int* __restrict__ ct_mask, int s7,
    const int* __restrict__ qm_plm,
    const int* __restrict__ tm_plm,
    const int* __restrict__ cm_plm, int plm_n,
    int* __restrict__ ws) {
    const int lane = threadIdx.x & (WAVE - 1);
    const int wv   = threadIdx.x >> 5;
    switch (wv) {
    case 0: wave_compact(qm_plm, plm_n, ws + WS_SRC_Q, lane); break;
    case 1: wave_compact(tm_plm, plm_n, ws + WS_SRC_T, lane); break;
    case 2: wave_compact(cm_plm, plm_n, ws + WS_SRC_C, lane); break;
    case 3: {
        int NQ = wave_popcount_mask(q_mask, s3, lane);
        int Sq = NQ > 0 ? s0 / NQ : 1;
        word_table(q_sub, NQ, Sq, ws + WS_L_Q, ws + WS_P_Q, lane);
        if (lane == 0) ws[WS_META + M_NQ] = NQ;
        break; }
    case 4: {
        int NT = wave_popcount_mask(tw_mask, s4, lane);
        int St = NT > 0 ? s1 / NT : 1;
        word_table(t_sub, NT, St, ws + WS_L_T, ws + WS_P_T, lane);
        if (lane == 0) ws[WS_META + M_NT] = NT;
        break; }
    case 5: {
        int NC = wave_popcount_mask(cw_mask, s5, lane);
        int Sc = NC > 0 ? s2 / NC : 1;
        word_table(c_sub, NC, Sc, ws + WS_L_C, ws + WS_P_C, lane);
        if (lane == 0) ws[WS_META + M_NC] = NC;
        break; }
    case 6:  wave_compact(q_mask,  s3, ws + WS_D_Q,  lane); break;
    case 7:  wave_compact(tw_mask, s4, ws + WS_D_TW, lane); break;
    case 8:  wave_compact(cw_mask, s5, ws + WS_D_CW, lane); break;
    case 9:  wave_compact(tt_mask, s6, ws + WS_D_TT, lane); break;
    case 10: wave_compact(ct_mask, s7, ws + WS_D_CT, lane); break;
    default: break;
    }
}

// ---------------- kernel 3: gather + mean-pool + scatter ----------------
// One wave per word. Source rows staged LDS via CDNA5 async global->LDS
// (ASYNCcnt path), pooled with float4 reads, scattered with b128 stores.
__global__ void __launch_bounds__(256)
SubAgg_pool_kernel(const float* __restrict__ x,
                   const int* __restrict__ ws,
                   float* __restrict__ out,
                   long oQ, long oTW, long oCW, long oTT, long oCT) {
    __shared__ float smem[8 * 3 * HDIM];   // 48 KB: 8 waves x 3 rows x 512 f32
    const int lane = threadIdx.x & (WAVE - 1);
    const int wv   = threadIdx.x >> 5;

    const int NQ = ws[WS_META + M_NQ];
    const int NT = ws[WS_META + M_NT];
    const int NC = ws[WS_META + M_NC];
    const int total = NQ + NT + NC;

    const int w = blockIdx.x * 8 + wv;
    if (w >= total) return;                 // wave-uniform exit

    const int *srcList, *Parr, *Larr, *dst0, *dst1;
    long ob0, ob1;
    int n;
    if (w < NQ) {
        n = w;
        srcList = ws + WS_SRC_Q; Parr = ws + WS_P_Q; Larr = ws + WS_L_Q;
        dst0 = ws + WS_D_Q;  ob0 = oQ;  dst1 = nullptr;       ob1 = 0;
    } else if (w < NQ + NT) {
        n = w - NQ;
        srcList = ws + WS_SRC_T; Parr = ws + WS_P_T; Larr = ws + WS_L_T;
        dst0 = ws + WS_D_TW; ob0 = oTW; dst1 = ws + WS_D_TT;  ob1 = oTT;
    } else {
        n = w - NQ - NT;
        srcList = ws + WS_SRC_C; Parr = ws + WS_P_C; Larr = ws + WS_L_C;
        dst0 = ws + WS_D_CW; ob0 = oCW; dst1 = ws + WS_D_CT;  ob1 = oCT;
    }

    const int P = Parr[n];
    const int L = Larr[n];
    if (L <= 0) return;
    const float inv = 1.0f / (float)L;

    float4 acc[4];
    #pragma unroll
    for (int j = 0; j < 4; ++j) { acc[j].x = acc[j].y = acc[j].z = acc[j].w = 0.f; }

    float* stage = smem + wv * 3 * HDIM;
    const uint32_t lds_base = (uint32_t)(uintptr_t)stage;   // low 32 bits = LDS offset

    for (int k0 = 0; k0 < L; k0 += 3) {
        const int kn = (L - k0 < 3) ? (L - k0) : 3;
        // --- async stage kn rows: per-lane 16B x 4, ASYNCcnt-tracked ---
        for (int k = 0; k < kn; ++k) {
            const float* row = x + (long)srcList[P + k0 + k] * HDIM;  // wave-uniform
            const uint32_t ldsRow = lds_base + (uint32_t)(k * HDIM * 4);
            #pragma unroll
            for (int j = 0; j < 4; ++j) {
                const uint32_t byteoff = (uint32_t)((lane + WAVE * j) * 16);
                asm volatile("global_load_async_to_lds_b128 %0, %1, %2"
                             :
                             : "v"(ldsRow + byteoff), "v"(byteoff), "s"(row)
                             : "memory");
            }
        }
        asm volatile("s_wait_asynccnt 0" ::: "memory");
        // --- accumulate from LDS (ds_load_b128) ---
        for (int k = 0; k < kn; ++k) {
            const float4* srow = (const float4*)(stage + k * HDIM);
            #pragma unroll
            for (int j = 0; j < 4; ++j) {
                float4 v = srow[lane + WAVE * j];
                acc[j].x += v.x * inv;
                acc[j].y += v.y * inv;
                acc[j].z += v.z * inv;
                acc[j].w += v.w * inv;
            }
        }
    }

    // --- scatter pooled row to destination(s), coalesced b128 stores ---
    float4* o0 = (float4*)(out + ob0 + (long)dst0[n] * HDIM);
    #pragma unroll
    for (int j = 0; j < 4; ++j) o0[lane + WAVE * j] = acc[j];
    if (dst1) {
        float4* o1 = (float4*)(out + ob1 + (long)dst1[n] * HDIM);
        #pragma unroll
        for (int j = 0; j < 4; ++j) o1[lane + WAVE * j] = acc[j];
    }
}

// ---------------- host launcher ----------------
extern "C" void kernel_launch(void* const* d_in, const int* in_sizes, int n_in,
                              void* d_out, int out_size, void* d_ws, size_t ws_size,
                              hipStream_t stream) {
    // setup_inputs() dict order:
    const int* q_sub   = (const int*)d_in[0];   // question_subword_mask
    const int* t_sub   = (const int*)d_in[1];   // table_subword_mask
    const int* c_sub   = (const int*)d_in[2];   // column_subword_mask
    const int* q_mask  = (const int*)d_in[3];   // question_mask
    const int* tw_mask = (const int*)d_in[4];   // table_word_mask
    const int* cw_mask = (const int*)d_in[5];   // column_word_mask
    const int* tt_mask = (const int*)d_in[6];   // table_total_mask
    const int* ct_mask = (const int*)d_in[7];   // column_total_mask
    const int* qm_plm  = (const int*)d_in[8];   // question_mask_plm  [B*512]
    const int* tm_plm  = (const int*)d_in[9];   // table_mask_plm
    const int* cm_plm  = (const int*)d_in[10];  // column_mask_plm
    const float* x     = (const float*)d_in[11];// inputs [B,512,512] f32
    float* out = (float*)d_out;
    int*   ws  = (int*)d_ws;                    // uses ~328 KB of scratch

    const int s0 = in_sizes[0], s1 = in_sizes[1], s2 = in_sizes[2];
    const int s3 = in_sizes[3], s4 = in_sizes[4], s5 = in_sizes[5];
    const int s6 = in_sizes[6], s7 = in_sizes[7];
    const int plm_n = in_sizes[8];

    // flat output offsets (floats), outputs concatenated in return order
    const long oQ  = 0;
    const long oTW = (long)s3 * HDIM;
    const long oCW = oTW + (long)s4 * HDIM;
    const long oTT = oCW + (long)s5 * HDIM;
    const long oCT = oTT + (long)s6 * HDIM;

    // 1) zero entire output (masked-off slots must be 0)
    long n4 = (long)out_size / 4;
    long zbl = (n4 + 255) / 256;
    int  zb  = (int)(zbl > 2048 ? 2048 : (zbl < 1 ? 1 : zbl));
    SubAgg_zero_kernel<<<zb, 256, 0, stream>>>((float4*)out, n4);

    // 2) build index tables (11 independent wave32 scan tasks)
    SubAgg_index_kernel<<<1, 11 * WAVE, 0, stream>>>(
        q_sub, s0, t_sub, s1, c_sub, s2,
        q_mask, s3, tw_mask, s4, cw_mask, s5,
        tt_mask, s6, ct_mask, s7,
        qm_plm, tm_plm, cm_plm, plm_n, ws);

    // 3) gather + pool + scatter; upper bound on word count is s3+s4+s5
    const int maxWords = s3 + s4 + s5;
    const int pb = (maxWords + 7) / 8;
    SubAgg_pool_kernel<<<pb, 256, 0, stream>>>(x, ws, out, oQ, oTW, oCW, oTT, oCT);
}